// EmbeddingLoss_89781996356333
// MI455X (gfx1250) — compile-verified
//
#include <hip/hip_runtime.h>

// Problem constants (match reference)
#define B_    8
#define K1_   19
#define K_    18          // keypoint groups used (k = 1..18)
#define LVEC_ 64
#define P_    16384       // H*W
#define NMAX_ 12
#define EPS_  1e-8f

typedef __attribute__((ext_vector_type(2))) float v2f;
typedef __attribute__((ext_vector_type(8))) float v8f;

// Workspace layout (floats):
//   [0, TVSZ)            tag_vec  [B][NMAX][K][LVEC]
//   [TVSZ, TVSZ+VSZ)     validf   [B][NMAX][K]
//   [TVSZ+VSZ, +B)       partial  [B]   (pull+push per batch)
//   [+B, +2B)            pcnt     [B]   (pair_count per batch)
#define TVSZ (B_*NMAX_*K_*LVEC_)
#define VSZ  (B_*NMAX_*K_)

__global__ void zero_ws(float* __restrict__ ws, int n) {
  int i = blockIdx.x * blockDim.x + threadIdx.x;
  int stride = gridDim.x * blockDim.x;
  for (; i < n; i += stride) ws[i] = 0.0f;
}

// One block per (b, k). Scan tags for nonzeros, gather 64-dim embeddings.
__global__ void __launch_bounds__(256)
gather_kernel(const float* __restrict__ inputs,
              const int*   __restrict__ tags,
              const int*   __restrict__ numH,
              float*       __restrict__ ws) {
  const int b   = blockIdx.x / K_;
  const int k   = blockIdx.x % K_;   // 0..17 (logical)
  const int kk  = k + 1;             // channel group in inputs/tags
  const int tid = threadIdx.x;
  const int nh  = numH[b];

  __shared__ int s_cnt;
  __shared__ int s_hit_n[16];
  __shared__ int s_hit_p[16];
  if (tid == 0) s_cnt = 0;
  __syncthreads();

  const int tbase = (b * K1_ + kk) * P_;
  for (int p = tid; p < P_; p += blockDim.x) {
    int t = tags[tbase + p];
    if (t > 0) {
      int n = t - 1;                 // person id 0..NMAX-1
      if (n < nh) {                  // fold person mask (matches reference valid)
        int idx = atomicAdd(&s_cnt, 1);
        if (idx < 16) { s_hit_n[idx] = n; s_hit_p[idx] = p; }
      }
    }
  }
  __syncthreads();
  const int cnt = (s_cnt < 16) ? s_cnt : 16;

  float* tv     = ws;
  float* validf = ws + TVSZ;
  // 4 hits serviced at a time: 64 lanes per hit gather the strided vector.
  for (int h0 = 0; h0 < cnt; h0 += 4) {
    int h    = h0 + (tid >> 6);
    int comp = tid & 63;
    if (h < cnt) {
      int n = s_hit_n[h];
      int p = s_hit_p[h];
      size_t src = ((size_t)b * (K1_ * LVEC_) + (size_t)(kk * LVEC_ + comp)) * P_ + p;
      tv[((b * NMAX_ + n) * K_ + k) * LVEC_ + comp] = inputs[src];
      if (comp == 0) validf[(b * NMAX_ + n) * K_ + k] = 1.0f;
    }
  }
}

// One block per batch: means, pull loss, and push loss (WMMA gram matrix).
__global__ void __launch_bounds__(256)
loss_kernel(const int* __restrict__ numH, float* __restrict__ ws) {
  const int b   = blockIdx.x;
  const int tid = threadIdx.x;
  int nh = numH[b];
  if (nh > NMAX_) nh = NMAX_;

  __shared__ float s_tv[NMAX_ * K_ * LVEC_];   // 55296 B
  __shared__ float s_valid[NMAX_ * K_];
  __shared__ float s_mean[NMAX_ * LVEC_];
  __shared__ float s_mhat[NMAX_ * LVEC_];
  __shared__ float s_mnorm[NMAX_];
  __shared__ float s_cntf[NMAX_];
  __shared__ float s_gram[16 * 16];
  __shared__ float s_red[256];

  const float* tv     = ws + (size_t)b * NMAX_ * K_ * LVEC_;
  const float* validf = ws + TVSZ + b * NMAX_ * K_;

  for (int i = tid; i < NMAX_ * K_ * LVEC_; i += 256) s_tv[i] = tv[i];
  for (int i = tid; i < NMAX_ * K_; i += 256) s_valid[i] = validf[i];
  __syncthreads();

  // counts per person
  if (tid < NMAX_) {
    float c = 0.f;
    for (int k = 0; k < K_; ++k) c += s_valid[tid * K_ + k];
    s_cntf[tid] = c;
  }
  __syncthreads();

  // mean[n][d] = sum_k tv / max(count,1)
  for (int i = tid; i < NMAX_ * LVEC_; i += 256) {
    int n = i / LVEC_, d = i % LVEC_;
    float s = 0.f;
    for (int k = 0; k < K_; ++k) s += s_tv[(n * K_ + k) * LVEC_ + d];
    s_mean[i] = s / fmaxf(s_cntf[n], 1.0f);
  }
  __syncthreads();

  // mnorm, mhat
  if (tid < NMAX_) {
    float sq = 0.f;
    for (int d = 0; d < LVEC_; ++d) { float v = s_mean[tid * LVEC_ + d]; sq += v * v; }
    float mn = fmaxf(sqrtf(sq), EPS_);
    s_mnorm[tid] = mn;
    float inv = 1.0f / mn;
    for (int d = 0; d < LVEC_; ++d) s_mhat[tid * LVEC_ + d] = s_mean[tid * LVEC_ + d] * inv;
  }
  __syncthreads();

  // pull loss: one (n,k) per thread (216 active)
  float pull_local = 0.f;
  if (tid < NMAX_ * K_) {
    int n = tid / K_;
    float dot = 0.f, sq = 0.f;
    for (int d = 0; d < LVEC_; ++d) {
      float v = s_tv[tid * LVEC_ + d];
      dot += v * s_mean[n * LVEC_ + d];
      sq  += v * v;
    }
    float tn   = fmaxf(sqrtf(sq), EPS_);
    float cosp = dot / (tn * s_mnorm[n]);
    pull_local = (1.0f - cosp) * s_valid[tid];
  }
  s_red[tid] = pull_local;
  __syncthreads();

  // push gram: G = mhat (12x64, zero-padded to 16) times its transpose,
  // via V_WMMA_F32_16X16X4_F32 accumulating over d in steps of 4.
  // f32 A(16x4) and B(4x16) share the lane layout K = vgpr + 2*(lane/16),
  // M=N=lane%16, so the symmetric gram uses the same registers for A and B.
  if (tid < 32) {
    int lane  = tid;
    int row   = lane & 15;
    int khalf = (lane >> 4) << 1;        // 0 or 2
    v8f c = {0.f, 0.f, 0.f, 0.f, 0.f, 0.f, 0.f, 0.f};
    for (int kb = 0; kb < LVEC_; kb += 4) {
      v2f a;
      if (row < NMAX_) {
        a.x = s_mhat[row * LVEC_ + kb + khalf + 0];
        a.y = s_mhat[row * LVEC_ + kb + khalf + 1];
      } else {
        a.x = 0.f; a.y = 0.f;
      }
      c = __builtin_amdgcn_wmma_f32_16x16x4_f32(
          /*neg_a=*/false, a, /*neg_b=*/false, a,
          /*c_mod=*/(short)0, c, /*reuse_a=*/false, /*reuse_b=*/false);
    }
    // C/D layout: VGPR r, lane L -> D[M = r + 8*(L/16)][N = L%16]
    for (int r = 0; r < 8; ++r) {
      int m = r + 8 * (lane >> 4);
      s_gram[m * 16 + row] = c[r];
    }
  }
  __syncthreads();

  if (tid == 0) {
    float pull = 0.f;
    for (int i = 0; i < 256; ++i) pull += s_red[i];
    float vsum = 0.f;
    for (int i = 0; i < NMAX_ * K_; ++i) vsum += s_valid[i];
    float push = 0.f, paircnt = 0.f;
    for (int n1 = 0; n1 < nh; ++n1)
      for (int n2 = 0; n2 < nh; ++n2)
        if (n1 != n2) {
          push += fmaxf(s_gram[n1 * 16 + n2], 0.0f);
          paircnt += 1.0f;
        }
    float* partial = ws + TVSZ + VSZ;
    float* pcnt    = partial + B_;
    partial[b] = pull + push;
    pcnt[b]    = vsum + paircnt;
  }
}

__global__ void finalize_kernel(const float* __restrict__ ws, float* __restrict__ out) {
  const float* partial = ws + TVSZ + VSZ;
  const float* pcnt    = partial + B_;
  float s = 0.f, pc = 0.f;
  for (int b = 0; b < B_; ++b) { s += partial[b]; pc += pcnt[b]; }
  float total = s / fmaxf(pc, 1.0f);
  out[0] = (pc > 0.f) ? total : 0.0f;
}

extern "C" void kernel_launch(void* const* d_in, const int* in_sizes, int n_in,
                              void* d_out, int out_size, void* d_ws, size_t ws_size,
                              hipStream_t stream) {
  const float* inputs = (const float*)d_in[0];
  const int*   tags   = (const int*)d_in[1];
  const int*   numH   = (const int*)d_in[2];
  float* ws  = (float*)d_ws;
  float* out = (float*)d_out;

  zero_ws<<<128, 256, 0, stream>>>(ws, TVSZ + VSZ);
  gather_kernel<<<B_ * K_, 256, 0, stream>>>(inputs, tags, numH, ws);
  loss_kernel<<<B_, 256, 0, stream>>>(numH, ws);
  finalize_kernel<<<1, 1, 0, stream>>>(ws, out);
}